// QuantumAttention_32950989095293
// MI455X (gfx1250) — compile-verified
//
#include <hip/hip_runtime.h>
#include <math.h>

// ---------------------------------------------------------------------------
// QuantumAttention on MI455X (gfx1250), fp32 WMMA path.
//
// Mathematically-exact simplification of the reference:
//   * softmax rows sum to 1 and the 'bhqk,bhvd->bhqd' contraction is
//     degenerate => attended[t, h*64+d] = sum_h' v_norm[t, h', d]  (q,k dead).
//   * o-projection of the head-tiled vector collapses to K=64 GEMMs against
//     Wc[o,d] = sum_h Wo[o,h*64+d]*cos(phase_o[h*64+d]) (Ws with sin).
// Dominant work: two fp32 GEMMs [16384x1024]x[1024x1024] (re/im of v),
// done with V_WMMA_F32_16X16X4_F32 for fp32 accuracy.
//
// Round-1 -> Round-2 change: 8 waves x 128 cols caused 128 accumulator VGPRs
// per wave and massive scratch spilling around the WMMAs. Now 16 waves x 64
// cols (64 accumulator VGPRs) + shuffle-tree norm reduction; no spills.
// ---------------------------------------------------------------------------

#define DIMN    1024
#define HD      64
#define TOKENS  (4 * 4096)
#define KT      32          // K-chunk staged in LDS
#define BLK_M   16          // tokens per workgroup
#define NTHREADS 512        // 16 wave32
#define WAVE_N  64          // output columns per wave (16 waves * 64 = 1024)

typedef float v2f __attribute__((ext_vector_type(2)));
typedef float v8f __attribute__((ext_vector_type(8)));

__device__ __forceinline__ v8f wmma4(v2f a, v2f b, v8f c) {
  // D = A(16x4 f32) * B(4x16 f32) + C(16x16 f32)
  return __builtin_amdgcn_wmma_f32_16x16x4_f32(false, a, false, b,
                                               (short)0, c, false, false);
}

// ---- LDS layout (floats) --------------------------------------------------
#define OFF_COS 0                       // 1024
#define OFF_SIN 1024                    // 1024
#define OFF_XS  2048                    // 16 rows * 34 (padded)    = 544
#define OFF_SRE 2592                    // 16 rows * 66 (padded)    = 1056
#define OFF_SIM 3648                    // 1056
#define OFF_N2  4704                    // 16
#define OFF_RN  4720                    // 16
#define OFF_RN2 4736                    // 16
#define OFF_B   4752                    // 16 waves * 64 rows * 34  = 34816
#define LDS_FLOATS (OFF_B + 16 * 64 * 34)   // 39568 floats = 158272 B (<320KB/WGP)

// ---------------------------------------------------------------------------
// Prep: fold Wo + phase_o into Wc/Ws  [1024 x 64] each.
// ---------------------------------------------------------------------------
__global__ void qa_prep(const float* __restrict__ Wo,
                        const float* __restrict__ pho,
                        float* __restrict__ Wc, float* __restrict__ Ws) {
  const int o = blockIdx.x;       // 0..1023
  const int d = threadIdx.x;      // 0..63
  float sc = 0.f, ss = 0.f;
  #pragma unroll
  for (int h = 0; h < 16; ++h) {
    const int j = h * HD + d;
    float s, c;
    sincosf(pho[j], &s, &c);
    const float w = Wo[(size_t)o * DIMN + j];
    sc += w * c;
    ss += w * s;
  }
  Wc[o * HD + d] = sc;
  Ws[o * HD + d] = ss;
}

// ---------------------------------------------------------------------------
// Fused main kernel: v-projection (WMMA) -> L2-normalize -> head-sum ->
// o-projection (WMMA, K=64) -> L2-normalize -> write complex64 output.
// Grid: TOKENS/16 blocks, 512 threads (16 wave32).
// ---------------------------------------------------------------------------
__global__ void __launch_bounds__(NTHREADS, 1)
qa_main(const float* __restrict__ x,
        const float* __restrict__ Wv,
        const float* __restrict__ bv,
        const float* __restrict__ phv,
        const float* __restrict__ bo,
        const float* __restrict__ Wc,
        const float* __restrict__ Ws,
        float* __restrict__ out) {
  extern __shared__ float sh[];
  const int tid  = threadIdx.x;
  const int lane = tid & 31;
  const int wid  = tid >> 5;          // wave id 0..15
  const int hf   = lane >> 4;         // lane half: 0 -> K+0, 1 -> K+2
  const int l16  = lane & 15;
  const int t0   = blockIdx.x * BLK_M;
  const int n0   = wid * WAVE_N;

  // --- init: per-channel cos/sin of phase_v; zero reduction buffers --------
  for (int i = tid; i < DIMN; i += NTHREADS) {
    float s, c;
    sincosf(phv[i], &s, &c);
    sh[OFF_COS + i] = c;
    sh[OFF_SIN + i] = s;
  }
  for (int i = tid; i < 16 * 66; i += NTHREADS) {
    sh[OFF_SRE + i] = 0.f;
    sh[OFF_SIM + i] = 0.f;
  }
  if (tid < 16) sh[OFF_N2 + tid] = 0.f;
  __syncthreads();

  v8f accRe[4], accIm[4];
  #pragma unroll
  for (int nt = 0; nt < 4; ++nt)
    #pragma unroll
    for (int j = 0; j < 8; ++j) { accRe[nt][j] = 0.f; accIm[nt][j] = 0.f; }

  // ======================= stage 1: v = (x.*e^{i phv}) @ Wv^T ==============
  for (int k0 = 0; k0 < DIMN; k0 += KT) {
    // stage x tile [16 x 32] (padded stride 34): 512 floats / 512 threads
    {
      const int m = tid >> 5, kk = tid & 31;
      sh[OFF_XS + m * 34 + kk] = x[(size_t)(t0 + m) * DIMN + k0 + kk];
    }
    // stage Wv tile: wave stages its 64 rows x 32 cols (padded stride 34)
    #pragma unroll
    for (int p = 0; p < 8; ++p) {
      const int r  = p * 8 + (lane >> 2);      // local row 0..63
      const int kc = (lane & 3) * 8;
      const float4* pw = (const float4*)(Wv + (size_t)(n0 + r) * DIMN + k0 + kc);
      const float4 w0 = pw[0], w1 = pw[1];
      float* dst = &sh[OFF_B + wid * 64 * 34 + r * 34 + kc];
      dst[0] = w0.x; dst[1] = w0.y; dst[2] = w0.z; dst[3] = w0.w;
      dst[4] = w1.x; dst[5] = w1.y; dst[6] = w1.z; dst[7] = w1.w;
    }
    // prefetch next chunk (global_prefetch_b8)
    if (k0 + KT < DIMN)
      __builtin_prefetch(Wv + (size_t)(n0 + (lane >> 2)) * DIMN + k0 + KT + (lane & 3) * 8);
    __syncthreads();

    #pragma unroll
    for (int kk = 0; kk < KT; kk += 4) {
      const int ks = kk + hf * 2;
      const v2f xa = *(const v2f*)&sh[OFF_XS + l16 * 34 + ks];
      const v2f cv = *(const v2f*)&sh[OFF_COS + k0 + ks];
      const v2f sv = *(const v2f*)&sh[OFF_SIN + k0 + ks];
      const v2f are = xa * cv;                 // A frag: x .* cos(phase)
      const v2f aim = xa * sv;                 // A frag: x .* sin(phase)
      const float* bb = &sh[OFF_B + wid * 64 * 34 + l16 * 34 + ks];
      #pragma unroll
      for (int nt = 0; nt < 4; ++nt) {
        const v2f bf = *(const v2f*)(bb + nt * 16 * 34);  // B[k][n] = Wv[n][k]
        accRe[nt] = wmma4(are, bf, accRe[nt]);
        accIm[nt] = wmma4(aim, bf, accIm[nt]);
      }
    }
    __syncthreads();
  }

  // ======================= stage 2: per-token L2 norm ======================
  // Each lane's 8 accumulator rows are tokens m = hf*8 + i; lanes with the
  // same half share the token set -> shuffle-reduce across the 16-lane half,
  // then one atomic per (wave, half, i).
  float part[8];
  #pragma unroll
  for (int j = 0; j < 8; ++j) part[j] = 0.f;
  #pragma unroll
  for (int nt = 0; nt < 4; ++nt) {
    const float bvn = bv[n0 + nt * 16 + l16];
    #pragma unroll
    for (int i = 0; i < 8; ++i) {
      accRe[nt][i] += bvn;                      // bias (zeros, kept faithful)
      part[i] += accRe[nt][i] * accRe[nt][i] + accIm[nt][i] * accIm[nt][i];
    }
  }
  #pragma unroll
  for (int i = 0; i < 8; ++i) {
    float p = part[i];
    #pragma unroll
    for (int ofs = 1; ofs < 16; ofs <<= 1) p += __shfl_xor(p, ofs, 32);
    if (l16 == 0) atomicAdd(&sh[OFF_N2 + hf * 8 + i], p);
  }
  __syncthreads();
  if (tid < 16) { sh[OFF_RN + tid] = rsqrtf(sh[OFF_N2 + tid]); sh[OFF_N2 + tid] = 0.f; }
  __syncthreads();

  // ======================= stage 3: head-sum of v_norm =====================
  // s[m][d] = sum_h v_norm[m][h*64+d]; this wave's 64 columns are head h=wid.
  #pragma unroll
  for (int nt = 0; nt < 4; ++nt) {
    const int d = nt * 16 + l16;               // n0 is a multiple of 64
    #pragma unroll
    for (int i = 0; i < 8; ++i) {
      const int m = hf * 8 + i;
      const float rn = sh[OFF_RN + m];
      atomicAdd(&sh[OFF_SRE + m * 66 + d], accRe[nt][i] * rn);
      atomicAdd(&sh[OFF_SIM + m * 66 + d], accIm[nt][i] * rn);
    }
  }
  __syncthreads();

  // ======================= stage 4: o-projection, K=64 =====================
  // y_re = s_re@Wc^T - s_im@Ws^T ; y_im = s_re@Ws^T + s_im@Wc^T
  v8f oRe[4], oIm[4];
  #pragma unroll
  for (int nt = 0; nt < 4; ++nt)
    #pragma unroll
    for (int j = 0; j < 8; ++j) { oRe[nt][j] = 0.f; oIm[nt][j] = 0.f; }

  #pragma unroll
  for (int kk = 0; kk < HD; kk += 4) {
    const int ks = kk + hf * 2;
    const v2f are  = *(const v2f*)&sh[OFF_SRE + l16 * 66 + ks];
    const v2f aim  = *(const v2f*)&sh[OFF_SIM + l16 * 66 + ks];
    const v2f aimn = -aim;   // f32 WMMA has no A-neg modifier; negate in VALU
    #pragma unroll
    for (int nt = 0; nt < 4; ++nt) {
      const size_t nrow = (size_t)(n0 + nt * 16 + l16) * HD + ks;
      const v2f wc = *(const v2f*)(Wc + nrow);   // hot in L2/WGP$ (512KB total)
      const v2f ws = *(const v2f*)(Ws + nrow);
      oRe[nt] = wmma4(are,  wc, oRe[nt]);
      oRe[nt] = wmma4(aimn, ws, oRe[nt]);
      oIm[nt] = wmma4(are,  ws, oIm[nt]);
      oIm[nt] = wmma4(aim,  wc, oIm[nt]);
    }
  }

  // ======================= stage 5: normalize + store complex64 ============
  float part2[8];
  #pragma unroll
  for (int j = 0; j < 8; ++j) part2[j] = 0.f;
  #pragma unroll
  for (int nt = 0; nt < 4; ++nt) {
    const float bon = bo[n0 + nt * 16 + l16];
    #pragma unroll
    for (int i = 0; i < 8; ++i) {
      oRe[nt][i] += bon;
      part2[i] += oRe[nt][i] * oRe[nt][i] + oIm[nt][i] * oIm[nt][i];
    }
  }
  #pragma unroll
  for (int i = 0; i < 8; ++i) {
    float p = part2[i];
    #pragma unroll
    for (int ofs = 1; ofs < 16; ofs <<= 1) p += __shfl_xor(p, ofs, 32);
    if (l16 == 0) atomicAdd(&sh[OFF_N2 + hf * 8 + i], p);
  }
  __syncthreads();
  if (tid < 16) sh[OFF_RN2 + tid] = rsqrtf(sh[OFF_N2 + tid]);
  __syncthreads();

  #pragma unroll
  for (int nt = 0; nt < 4; ++nt) {
    const int n = n0 + nt * 16 + l16;
    #pragma unroll
    for (int i = 0; i < 8; ++i) {
      const int m = hf * 8 + i;
      const float rn = sh[OFF_RN2 + m];
      float2 cv = make_float2(oRe[nt][i] * rn, oIm[nt][i] * rn);
      *(float2*)(out + ((size_t)(t0 + m) * DIMN + n) * 2) = cv;  // complex64
    }
  }
}

// ---------------------------------------------------------------------------
extern "C" void kernel_launch(void* const* d_in, const int* in_sizes, int n_in,
                              void* d_out, int out_size, void* d_ws, size_t ws_size,
                              hipStream_t stream) {
  (void)in_sizes; (void)n_in; (void)out_size; (void)ws_size;
  // setup_inputs order: x, Wq,bq,phq, Wk,bk,phk, Wv,bv,phv, Wo,bo,pho
  const float* x   = (const float*)d_in[0];
  const float* Wv  = (const float*)d_in[7];
  const float* bv  = (const float*)d_in[8];
  const float* phv = (const float*)d_in[9];
  const float* Wo  = (const float*)d_in[10];
  const float* bo  = (const float*)d_in[11];
  const float* pho = (const float*)d_in[12];

  float* Wc = (float*)d_ws;                 // [1024 x 64]
  float* Ws = Wc + DIMN * HD;               // [1024 x 64]  (512 KB total)
  float* out = (float*)d_out;               // complex64 as interleaved floats

  qa_prep<<<DIMN, HD, 0, stream>>>(Wo, pho, Wc, Ws);

  const size_t shbytes = (size_t)LDS_FLOATS * sizeof(float);
  (void)hipFuncSetAttribute((const void*)qa_main,
                            hipFuncAttributeMaxDynamicSharedMemorySize,
                            (int)shbytes);
  qa_main<<<TOKENS / BLK_M, NTHREADS, shbytes, stream>>>(x, Wv, bv, phv, bo, Wc, Ws, out);
}